// CustomContrastiveLoss_71811853189567
// MI455X (gfx1250) — compile-verified
//
#include <hip/hip_runtime.h>
#include <stdint.h>

typedef float v2f __attribute__((ext_vector_type(2)));
typedef float v8f __attribute__((ext_vector_type(8)));

#define N_TOT   8192
#define DDIM    128
#define CSPLIT  8                     // column chunks (grid parallelism)
#define CHUNK   (N_TOT / CSPLIT)      // 1024 columns per block
#define TILE    64                    // columns staged per LDS buffer
#define NTILES  (CHUNK / TILE)        // 16
#define LDS_STRIDE 132                // 128 + 4 pad -> conflict-free b64 frags
#define LOG2E    1.4426950408889634f
#define CLIP_VAL 39.8631371f          // -log2(float(1e-12))

// CDNA5 async memory->LDS copy (ASYNCcnt-tracked, no VGPR destination).
__device__ __forceinline__ void async_copy_b128(uint32_t lds_off,
                                                const void* gsrc) {
  asm volatile("global_load_async_to_lds_b128 %0, %1, off"
               :: "v"(lds_off), "v"(gsrc) : "memory");
}
__device__ __forceinline__ void wait_asynccnt0() {
  asm volatile("s_wait_asynccnt 0x0" ::: "memory");
}

// ---------------------------------------------------------------------------
// Pass 1: flash-style partial row statistics (base-2 online softmax).
// Block = 8 waves x 16 rows = 128 rows over a 1024-column chunk.
// B tiles (64 cols x 128 K) double-buffered in LDS via async-to-LDS DMA;
// per-lane online (m,l); one 16-lane merge at the end.
// ---------------------------------------------------------------------------
__global__ __launch_bounds__(256) void pass1_rowstats(
    const float* __restrict__ logits, const float* __restrict__ labels,
    const int* __restrict__ pad_mask,
    float* __restrict__ pM, float* __restrict__ pL)
{
  __shared__ float smem[2][TILE * LDS_STRIDE];

  const int tid  = threadIdx.x;
  const int lane = tid & 31;
  const int wave = tid >> 5;
  const int rb   = (int)blockIdx.x >> 3;            // row block 0..63
  const int cs   = (int)blockIdx.x & (CSPLIT - 1);  // column chunk 0..7
  const int rowBase   = rb * 128 + wave * 16;
  const int chunkBase = cs * CHUNK;
  const int r0 = lane & 15;                         // row (A) / col (B) in tile
  const int hi = lane >> 4;                         // K-pair selector

  // A fragments: 16 rows x 128 K (WMMA f32 16x16x4 layout), resident in VGPRs.
  v2f a[32];
  {
    const float* Ap = logits + (size_t)(rowBase + r0) * DDIM + 2 * hi;
#pragma unroll
    for (int kk = 0; kk < 32; ++kk)
      a[kk] = *reinterpret_cast<const v2f*>(Ap + 4 * kk);
  }

  float m[8], l[8];
#pragma unroll
  for (int r = 0; r < 8; ++r) { m[r] = -INFINITY; l[r] = 0.0f; }

  // 64 rows x 32 float4 per tile; 8 async b128 per thread, no staging VGPRs.
  auto async_load_tile = [&](int c0, int buf) {
#pragma unroll
    for (int i = 0; i < 8; ++i) {
      const int idx = tid + 256 * i;            // 0..2047
      const int row = idx >> 5, c4 = idx & 31;
      const float* src = labels + (size_t)(c0 + row) * DDIM + c4 * 4;
      const uint32_t dst =
          (uint32_t)(uintptr_t)&smem[buf][row * LDS_STRIDE + c4 * 4];
      async_copy_b128(dst, src);
      if ((c4 & 7) == 0 && c0 + 2 * TILE <= chunkBase + CHUNK - TILE) {
        // one speculative prefetch per 128B line, two tiles ahead
        __builtin_prefetch(
            labels + (size_t)(c0 + 2 * TILE + row) * DDIM + c4 * 4, 0, 1);
      }
    }
  };

  async_load_tile(chunkBase, 0);
  wait_asynccnt0();
  __syncthreads();

  for (int tt = 0; tt < NTILES; ++tt) {
    const int cur = tt & 1;
    if (tt + 1 < NTILES)
      async_load_tile(chunkBase + (tt + 1) * TILE, (tt + 1) & 1);

#pragma unroll
    for (int st = 0; st < 4; ++st) {                 // four 16-col subtiles
      const int col = chunkBase + tt * TILE + st * 16 + r0;
      const float* bp = &smem[cur][(st * 16 + r0) * LDS_STRIDE + 2 * hi];

      v8f c;
#pragma unroll
      for (int r = 0; r < 8; ++r) c[r] = 0.0f;

      v2f bq[4];
#pragma unroll
      for (int p = 0; p < 4; ++p)
        bq[p] = *reinterpret_cast<const v2f*>(bp + 4 * p);
#pragma unroll
      for (int kk = 0; kk < 32; ++kk) {
        const v2f bc = bq[kk & 3];
        if (kk + 4 < 32)
          bq[kk & 3] = *reinterpret_cast<const v2f*>(bp + 4 * (kk + 4));
        c = __builtin_amdgcn_wmma_f32_16x16x4_f32(false, a[kk], false, bc,
                                                  (short)0, c, false, false);
      }

      const float cadd = pad_mask[col] ? 0.0f : -INFINITY;
#pragma unroll
      for (int r = 0; r < 8; ++r) {
        const float s2 = c[r] * LOG2E + cadd;        // base-2 logit
        if (s2 > m[r]) {                             // rare after warmup
          l[r] = l[r] * __builtin_amdgcn_exp2f(m[r] - s2) + 1.0f;
          m[r] = s2;
        } else if (s2 != -INFINITY) {
          l[r] += __builtin_amdgcn_exp2f(s2 - m[r]);
        }
      }
    }

    wait_asynccnt0();   // own async issues for tile t+1 landed in LDS
    __syncthreads();    // everyone done reading buf cur / writing buf next
  }

  // ---- one-time 16-lane merge (masks 1..8 stay inside each half) ----
#pragma unroll
  for (int r = 0; r < 8; ++r) {
    float mr = m[r], lr = l[r];
#pragma unroll
    for (int off = 1; off < 16; off <<= 1) {
      const float om = __shfl_xor(mr, off, 32);
      const float ol = __shfl_xor(lr, off, 32);
      const float nm = fmaxf(mr, om);
      const float e1 = (mr == -INFINITY) ? 0.0f : __builtin_amdgcn_exp2f(mr - nm);
      const float e2 = (om == -INFINITY) ? 0.0f : __builtin_amdgcn_exp2f(om - nm);
      lr = lr * e1 + ol * e2;
      mr = nm;
    }
    if (r0 == 0) {
      const int row = rowBase + 8 * hi + r;
      pM[cs * N_TOT + row] = mr;
      pL[cs * N_TOT + row] = lr;
    }
  }
}

// ---------------------------------------------------------------------------
// Pass 1b: merge the CSPLIT partial (m,l) per row -> M2_i = log2(Z_i) + m_i
// ---------------------------------------------------------------------------
__global__ __launch_bounds__(256) void pass1_merge(
    const float* __restrict__ pM, const float* __restrict__ pL,
    float* __restrict__ Mvals)
{
  const int row = (int)(blockIdx.x * blockDim.x + threadIdx.x);
  if (row >= N_TOT) return;
  float M = -INFINITY;
#pragma unroll
  for (int c = 0; c < CSPLIT; ++c) M = fmaxf(M, pM[c * N_TOT + row]);
  float L = 0.0f;
#pragma unroll
  for (int c = 0; c < CSPLIT; ++c) {
    const float mc = pM[c * N_TOT + row];
    if (mc != -INFINITY) L += pL[c * N_TOT + row] * __builtin_amdgcn_exp2f(mc - M);
  }
  Mvals[row] = M + log2f(L);
}

// ---------------------------------------------------------------------------
// Pass 2: sparse positive-pair loss (ballot-scan of ad[], ~8 dots per row).
// loss_i = sum_{j: ad_j==ad_i, valid} min(M2_i - s_ij*log2e, CLIP)
// ---------------------------------------------------------------------------
__global__ __launch_bounds__(256) void pass2_posloss(
    const float* __restrict__ logits, const float* __restrict__ labels,
    const int* __restrict__ pad_mask, const long long* __restrict__ ad,
    const float* __restrict__ Mvals, float* __restrict__ rowloss)
{
  const int lane = threadIdx.x & 31;
  const int i    = (int)((blockIdx.x * blockDim.x + threadIdx.x) >> 5);
  if (i >= N_TOT) return;

  const long long adi = ad[i];
  const int   vi = pad_mask[i];
  const float Mi = Mvals[i];
  const float4 q =
      *reinterpret_cast<const float4*>(logits + (size_t)i * DDIM + lane * 4);

  float loss = 0.0f;
  for (int jb = 0; jb < N_TOT; jb += 32) {
    const int  j     = jb + lane;
    const bool match = vi && pad_mask[j] && (ad[j] == adi);
    unsigned long long mb = __ballot(match);
    while (mb) {
      const int jj = jb + (__ffsll(mb) - 1);
      mb &= mb - 1;
      const float4 kv = *reinterpret_cast<const float4*>(
          labels + (size_t)jj * DDIM + lane * 4);
      float p = q.x * kv.x + q.y * kv.y + q.z * kv.z + q.w * kv.w;
#pragma unroll
      for (int off = 1; off < 32; off <<= 1)
        p += __shfl_xor(p, off, 32);
      loss += fminf(Mi - p * LOG2E, CLIP_VAL);
    }
  }
  if (lane == 0) rowloss[i] = loss;
}

// ---------------------------------------------------------------------------
// Pass 3: mean over rows.
// ---------------------------------------------------------------------------
__global__ __launch_bounds__(256) void pass3_reduce(
    const float* __restrict__ rowloss, float* __restrict__ out)
{
  __shared__ float sm[256];
  float s = 0.0f;
  for (int i = threadIdx.x; i < N_TOT; i += 256) s += rowloss[i];
  sm[threadIdx.x] = s;
  __syncthreads();
  for (int stride = 128; stride > 0; stride >>= 1) {
    if ((int)threadIdx.x < stride) sm[threadIdx.x] += sm[threadIdx.x + stride];
    __syncthreads();
  }
  if (threadIdx.x == 0) out[0] = sm[0] / (float)N_TOT;
}

extern "C" void kernel_launch(void* const* d_in, const int* in_sizes, int n_in,
                              void* d_out, int out_size, void* d_ws, size_t ws_size,
                              hipStream_t stream) {
  (void)in_sizes; (void)n_in; (void)out_size; (void)ws_size;
  const float*     logits   = (const float*)d_in[0];
  const float*     labels   = (const float*)d_in[1];
  const int*       pad_mask = (const int*)d_in[2];
  const long long* ad       = (const long long*)d_in[3];
  float* out = (float*)d_out;

  float* pM      = (float*)d_ws;                 // CSPLIT * N
  float* pL      = pM + (size_t)CSPLIT * N_TOT;  // CSPLIT * N
  float* Mvals   = pL + (size_t)CSPLIT * N_TOT;  // N
  float* rowloss = Mvals + N_TOT;                // N

  pass1_rowstats<<<64 * CSPLIT, 256, 0, stream>>>(logits, labels, pad_mask,
                                                  pM, pL);
  pass1_merge<<<N_TOT / 256, 256, 0, stream>>>(pM, pL, Mvals);
  pass2_posloss<<<1024, 256, 0, stream>>>(logits, labels, pad_mask, ad,
                                          Mvals, rowloss);
  pass3_reduce<<<1, 256, 0, stream>>>(rowloss, out);
}